// SematicVoxelization_32057635897982
// MI455X (gfx1250) — compile-verified
//
#include <hip/hip_runtime.h>

// ---------------------------------------------------------------------------
// Semantic voxelization (PaMIR-style Gaussian splat), MI455X / gfx1250.
//   sem  : 256^3 x 3 floats  (d_out[0 .. 50331647])
//   wsum : 256^3 floats      (d_out[50331648 .. 67108863]), init 0.001
// w = exp(-d2/(2 sig^2)) = 2^(E*d2), E = -8192*log2(e).  Constants are
// pre-folded into the operands:
//   E*d2 = (E*vs^2*|off|^2)[m] + (off . (2*vs*E*t))[m,n] + (E*|t|^2)[n]
// The middle (cross) term for a 16-offset x 16-vertex tile is computed by
// V_WMMA_F32_16X16X4_F32 with pre-scaled t-columns in B.  Tail per element:
// 2 adds + v_exp_f32 + gated global_atomic_add_f32 scatter.
// ---------------------------------------------------------------------------

#define RES       256
#define NV        6890
#define K_OFF     343           // 7^3 window
#define NPAD      352           // 22 tiles * 16
#define NTILE     22
#define VS        (1.0f/256.0f) // voxel size, H=1
// E = -INV2SIG2 * log2(e) = -8192 * 1.4426950408889634
#define E2N       (-11818.5577749632f)
#define O2SCALE   (E2N * VS * VS)      // scales |off|^2 table
#define BSCALE    (2.0f * VS * E2N)    // scales t columns fed to WMMA

typedef float v2f __attribute__((ext_vector_type(2)));
typedef float v8f __attribute__((ext_vector_type(8)));

// ---- output initialization: 256MB streaming fill, float4 stores -----------
__global__ void sv_init_out(float4* __restrict__ out4) {
  const long long totN4 = ((long long)RES * RES * RES * 4) >> 2; // 16,777,216
  const long long sem4  = ((long long)RES * RES * RES * 3) >> 2; // 12,582,912
  long long i      = (long long)blockIdx.x * blockDim.x + threadIdx.x;
  long long stride = (long long)gridDim.x * blockDim.x;
  for (long long j = i; j < totN4; j += stride) {
    float v = (j >= sem4) ? 0.001f : 0.0f;   // boundary is float4-aligned
    out4[j] = make_float4(v, v, v, v);
  }
}

// ---- splat kernel: 1 block = 16 vertices; 4 waves x 22 offset tiles -------
__global__ __launch_bounds__(128)
void sv_splat(const float* __restrict__ verts,
              const float* __restrict__ codes,
              const float* __restrict__ occ,
              float* __restrict__ sem,
              float* __restrict__ wsum) {
  __shared__ int   s_cx[16], s_cy[16], s_cz[16], s_val[16];  // centers - 3
  __shared__ float s_tx[16], s_ty[16], s_tz[16];             // t * BSCALE
  __shared__ float s_t2e[16];                                // |t|^2 * E2N
  __shared__ float s_c0[16], s_c1[16], s_c2[16];
  __shared__ unsigned s_off[NPAD];   // packed dx | dy<<8 | dz<<16, in [0,6]
  __shared__ float    s_o2e[NPAD];   // |off|^2 * O2SCALE

  const int tid = threadIdx.x;

  // offset decode table (built once per block, replaces div/mod chains)
  for (int i = tid; i < NPAD; i += 128) {
    int m  = (i < K_OFF) ? i : 0;
    int dx = m / 49;
    int rm = m - dx * 49;
    int dy = rm / 7;
    int dz = rm - dy * 7;              // all in [0,6]
    s_off[i] = (unsigned)(dx | (dy << 8) | (dz << 16));
    int ax = dx - 3, ay = dy - 3, az = dz - 3;
    s_o2e[i] = (float)(ax * ax + ay * ay + az * az) * O2SCALE;
  }

  if (tid < 16) {
    int v     = blockIdx.x * 16 + tid;
    int valid = (v < NV);
    int vi    = valid ? v : 0;
    float vx = verts[vi * 3 + 0], vy = verts[vi * 3 + 1], vz = verts[vi * 3 + 2];
    int cx = (int)floorf((vx + 0.5f) * 256.0f);
    int cy = (int)floorf((vy + 0.5f) * 256.0f);
    int cz = (int)floorf((vz + 0.5f) * 256.0f);
    float tx = ((float)cx + 0.5f) * VS - 0.5f - vx;
    float ty = ((float)cy + 0.5f) * VS - 0.5f - vy;
    float tz = ((float)cz + 0.5f) * VS - 0.5f - vz;
    s_cx[tid] = cx - 3; s_cy[tid] = cy - 3; s_cz[tid] = cz - 3;
    s_tx[tid] = tx * BSCALE;
    s_ty[tid] = ty * BSCALE;
    s_tz[tid] = tz * BSCALE;
    s_t2e[tid] = (tx * tx + ty * ty + tz * tz) * E2N;
    s_c0[tid] = codes[vi * 3 + 0];
    s_c1[tid] = codes[vi * 3 + 1];
    s_c2[tid] = codes[vi * 3 + 2];
    s_val[tid] = valid;
  }
  __syncthreads();

  const int wave = tid >> 5;        // 0..3
  const int lane = tid & 31;
  const int half = lane >> 4;       // 0: K=0,1 ; 1: K=2,3 (A/B VGPR layout)
  const int l16  = lane & 15;

  // B operand (4x16, pre-scaled t-columns): lanes 0-15 -> {tx,ty}, 16-31 -> {tz,0}
  v2f B;
  B[0] = half ? s_tz[l16] : s_tx[l16];
  B[1] = half ? 0.0f      : s_ty[l16];

  // per-lane vertex-N broadcast state (N = lane & 15 in C/D layout)
  const float t2e = s_t2e[l16];
  const int   cx = s_cx[l16], cy = s_cy[l16], cz = s_cz[l16];  // pre-biased -3
  const float c0 = s_c0[l16], c1 = s_c1[l16], c2 = s_c2[l16];
  const int   vvalid = s_val[l16];

  for (int tile = wave; tile < NTILE; tile += 4) {
    // A operand (16x4, offset rows): row M = l16 of this tile (table lookup)
    unsigned pa = s_off[tile * 16 + l16];
    float ox = (float)(pa & 255u)         - 3.0f;
    float oy = (float)((pa >> 8) & 255u)  - 3.0f;
    float oz = (float)((pa >> 16) & 255u) - 3.0f;
    v2f A;
    A[0] = half ? oz   : ox;
    A[1] = half ? 0.0f : oy;

    // P[m,n] = off_m . (2*vs*E * t_n)  -> middle term of E*d2 directly
    v8f P = {};
    P = __builtin_amdgcn_wmma_f32_16x16x4_f32(
        /*neg_a=*/false, A, /*neg_b=*/false, B,
        /*c_mod=*/(short)0, P, /*reuse_a=*/false, /*reuse_b=*/false);

    // ---- phase 1: batch all 8 occ gathers (wrapped index, mask later) ----
    int   lin[8];
    float occv[8], o2e[8];
    int   ok[8];
#pragma unroll
    for (int r = 0; r < 8; ++r) {
      int m = tile * 16 + r + half * 8;         // broadcast read within half
      unsigned p = s_off[m];
      int ix = cx + (int)(p & 255u);
      int iy = cy + (int)((p >> 8) & 255u);
      int iz = cz + (int)((p >> 16) & 255u);
      ok[r] = ((unsigned)ix < (unsigned)RES) & ((unsigned)iy < (unsigned)RES) &
              ((unsigned)iz < (unsigned)RES) & (m < K_OFF) & vvalid;
      lin[r]  = (((ix & 255) << 8 | (iy & 255)) << 8) | (iz & 255);
      o2e[r]  = s_o2e[m];
      occv[r] = occ[lin[r]];                    // always in-range (wrapped)
    }
    // ---- phase 2: gate, exp2, scatter-accumulate -------------------------
#pragma unroll
    for (int r = 0; r < 8; ++r) {
      if (ok[r] && occv[r] > 0.0f) {
        float xe = o2e[r] + P[r] + t2e;         // = E * d2
        float w  = __builtin_amdgcn_exp2f(xe);  // v_exp_f32 (base-2)
        atomicAdd(&sem[lin[r] * 3 + 0], w * c0);
        atomicAdd(&sem[lin[r] * 3 + 1], w * c1);
        atomicAdd(&sem[lin[r] * 3 + 2], w * c2);
        atomicAdd(&wsum[lin[r]], w);
      }
    }
  }
}

extern "C" void kernel_launch(void* const* d_in, const int* in_sizes, int n_in,
                              void* d_out, int out_size, void* d_ws, size_t ws_size,
                              hipStream_t stream) {
  const float* verts = (const float*)d_in[0];   // (6890,3)
  const float* codes = (const float*)d_in[1];   // (6890,3)
  const float* occ   = (const float*)d_in[2];   // (256,256,256)
  (void)d_in[3];                                // smpl_faces: unused by reference
  (void)in_sizes; (void)n_in; (void)out_size; (void)d_ws; (void)ws_size;

  float* out  = (float*)d_out;
  float* sem  = out;                                        // 256^3 * 3
  float* wsum = out + (long long)RES * RES * RES * 3;       // 256^3

  // 1) initialize sem=0, wsum=0.001 (256MB streaming, HBM-bound floor)
  sv_init_out<<<8192, 256, 0, stream>>>((float4*)out);

  // 2) WMMA-tiled Gaussian splat with L2 atomics
  int vgroups = (NV + 15) / 16;                             // 431 blocks
  sv_splat<<<vgroups, 128, 0, stream>>>(verts, codes, occ, sem, wsum);
}